// NodeGCN_5634997092607
// MI455X (gfx1250) — compile-verified
//
#include <hip/hip_runtime.h>

typedef __attribute__((ext_vector_type(2))) float v2f;
typedef __attribute__((ext_vector_type(8))) float v8f;

#define HD 20
#define HDPAD 32          // HD padded to 2 WMMA col tiles
#define CD 10
#define CDPAD 16          // CD padded to 1 WMMA col tile
#define BN_EPS 1e-5f

// ---------------------------------------------------------------- utilities
__global__ void fill_zero_kernel(float* __restrict__ p, int n) {
  int i = blockIdx.x * blockDim.x + threadIdx.x;
  int st = gridDim.x * blockDim.x;
  for (; i < n; i += st) p[i] = 0.0f;
}

// Pack W[K,Hd] (row-major) into zero-padded transposed Wt[HdPad][K]:
// Wt[c*K + k] = (c < Hd) ? W[k*Hd + c] : 0
__global__ void pack_w_kernel(const float* __restrict__ W, float* __restrict__ Wt,
                              int K, int Hd, int HdPad) {
  int t = blockIdx.x * blockDim.x + threadIdx.x;
  if (t < HdPad * K) {
    int c = t / K, k = t - c * K;
    Wt[t] = (c < Hd) ? W[k * Hd + c] : 0.0f;
  }
}

// deg[dst] += ew  (edge part of segment_sum for degree)
__global__ void deg_kernel(const int* __restrict__ dst, const float* __restrict__ ew,
                           float* __restrict__ deg, int E) {
  int i = blockIdx.x * blockDim.x + threadIdx.x;
  if (i < E) unsafeAtomicAdd(&deg[dst[i]], ew[i]);
}

// dinv = rsqrt(deg + 1) (self-loop weight 1), in place
__global__ void dinv_kernel(float* __restrict__ deg, int N) {
  int i = blockIdx.x * blockDim.x + threadIdx.x;
  if (i < N) {
    float d = deg[i] + 1.0f;
    deg[i] = (d > 0.0f) ? rsqrtf(d) : 0.0f;
  }
}

// ------------------------------------------------- WMMA fp32 GEMM: Out = X@W
// X: [N,K] row-major; Wt: [HdPad][K] transposed + zero-padded; Out: [N,Hd].
// N % 16 == 0, K % 4 == 0. One wave = one 16x16 tile of V_WMMA_F32_16X16X4_F32.
// No predication in the K loop: padded columns contribute exact zeros.
__global__ __launch_bounds__(256)
void gemm_wmma_kernel(const float* __restrict__ X, const float* __restrict__ Wt,
                      float* __restrict__ Out, int N, int K, int Hd, int HdPad) {
  int nRowTiles = N >> 4;
  int nColTiles = HdPad >> 4;
  int wave = (blockIdx.x * blockDim.x + threadIdx.x) >> 5;
  if (wave >= nRowTiles * nColTiles) return;          // wave-uniform exit
  int lane = threadIdx.x & 31;
  int l16  = lane & 15;
  int half = lane >> 4;                               // 0: K pair {k,k+1}, 1: {k+2,k+3}
  int rowTile = wave % nRowTiles;
  int colTile = wave / nRowTiles;
  int row = (rowTile << 4) + l16;                     // A: row per lane
  int col = (colTile << 4) + l16;                     // B/D: col per lane (< HdPad)
  const float* xrow = X  + (size_t)row * K + 2 * half;
  const float* wcol = Wt + (size_t)col * K + 2 * half;
  v8f acc = {};
  for (int k = 0; k < K; k += 4) {
    v2f a = *(const v2f*)(xrow + k);                  // 8B-aligned global_load_b64
    v2f b = *(const v2f*)(wcol + k);                  // contiguous K pair, unguarded
    acc = __builtin_amdgcn_wmma_f32_16x16x4_f32(false, a, false, b,
                                                (short)0, acc, false, false);
  }
  if (col < Hd) {
    int base = ((rowTile << 4) + (half << 3)) * Hd + col;  // D: VGPR v -> row v+8*half
    #pragma unroll
    for (int v = 0; v < 8; v++) Out[base + v * Hd] = acc[v];
  }
}

// ------------------------------------------ edge scatter: out[d] += h[s]*coef
__global__ void edge_scatter_kernel(const int* __restrict__ src, const int* __restrict__ dst,
                                    const float* __restrict__ ew, const float* __restrict__ dinv,
                                    const float* __restrict__ h, float* __restrict__ out, int E) {
  int i = blockIdx.x * blockDim.x + threadIdx.x;
  if (i >= E) return;
  int s = src[i], d = dst[i];
  float c = dinv[s] * ew[i] * dinv[d];
  const float4* hs = (const float4*)(h + (size_t)s * HD);  // 80B rows -> 16B aligned
  float* od = out + (size_t)d * HD;
  #pragma unroll
  for (int q = 0; q < 5; q++) {
    float4 r = hs[q];
    unsafeAtomicAdd(od + 4 * q + 0, r.x * c);
    unsafeAtomicAdd(od + 4 * q + 1, r.y * c);
    unsafeAtomicAdd(od + 4 * q + 2, r.z * c);
    unsafeAtomicAdd(od + 4 * q + 3, r.w * c);
  }
}

// ------------ self-loop + bias + ReLU, optional BN stats (sum / sumsq per feature)
__global__ __launch_bounds__(256)
void post_kernel(const float* __restrict__ conv, const float* __restrict__ h,
                 const float* __restrict__ dinv, const float* __restrict__ bias,
                 float* __restrict__ o, float* __restrict__ stats, int N, int do_stats) {
  __shared__ float ssum[HD], ssq[HD];
  if (do_stats && threadIdx.x < HD) { ssum[threadIdx.x] = 0.0f; ssq[threadIdx.x] = 0.0f; }
  __syncthreads();
  int i = blockIdx.x * blockDim.x + threadIdx.x;
  int lane = threadIdx.x & 31;
  bool act = i < N;
  float sl = act ? dinv[i] * dinv[i] : 0.0f;          // self-loop coef (ew=1)
  #pragma unroll
  for (int j = 0; j < HD; j++) {
    float v = 0.0f;
    if (act) {
      int idx = i * HD + j;
      v = conv[idx] + h[idx] * sl + bias[j];
      v = v > 0.0f ? v : 0.0f;                        // ReLU
      o[idx] = v;
    }
    if (do_stats) {
      float s1 = v, s2 = v * v;
      #pragma unroll
      for (int off = 16; off > 0; off >>= 1) {        // wave32 reduction
        s1 += __shfl_down(s1, off, 32);
        s2 += __shfl_down(s2, off, 32);
      }
      if (lane == 0) { atomicAdd(&ssum[j], s1); atomicAdd(&ssq[j], s2); }
    }
  }
  if (do_stats) {
    __syncthreads();
    if (threadIdx.x < HD) {
      unsafeAtomicAdd(&stats[threadIdx.x], ssum[threadIdx.x]);
      unsafeAtomicAdd(&stats[HD + threadIdx.x], ssq[threadIdx.x]);
    }
  }
}

// stats[0..19]=sum, [20..39]=sumsq  ->  [40..59]=scale, [60..79]=shift
__global__ void bn_finalize_kernel(float* __restrict__ stats, const float* __restrict__ g,
                                   const float* __restrict__ be, float invN) {
  int j = threadIdx.x;
  if (j < HD) {
    float m = stats[j] * invN;
    float var = stats[HD + j] * invN - m * m;
    float sc = g[j] * rsqrtf(var + BN_EPS);
    stats[2 * HD + j] = sc;
    stats[3 * HD + j] = be[j] - m * sc;
  }
}

__global__ void bn_apply_kernel(float* __restrict__ o, const float* __restrict__ stats, int n) {
  int i = blockIdx.x * blockDim.x + threadIdx.x;
  int st = gridDim.x * blockDim.x;
  for (; i < n; i += st) {
    int j = i % HD;
    o[i] = o[i] * stats[2 * HD + j] + stats[3 * HD + j];
  }
}

// ------------- classifier: out[N,10] = concat(o1,o2,o3)[N,60] @ Wl[60,10] + bl
// Wtl: [CDPAD][60] transposed + zero-padded classifier weights.
__global__ __launch_bounds__(256)
void final_gemm_kernel(const float* __restrict__ o1, const float* __restrict__ o2,
                       const float* __restrict__ o3, const float* __restrict__ Wtl,
                       const float* __restrict__ bl, float* __restrict__ out, int N) {
  int nRowTiles = N >> 4;
  int wave = (blockIdx.x * blockDim.x + threadIdx.x) >> 5;
  if (wave >= nRowTiles) return;                      // wave-uniform exit
  int lane = threadIdx.x & 31;
  int l16  = lane & 15;
  int half = lane >> 4;
  int row = (wave << 4) + l16;
  const float* r1 = o1 + (size_t)row * HD + 2 * half;
  const float* r2 = o2 + (size_t)row * HD - 20 + 2 * half;
  const float* r3 = o3 + (size_t)row * HD - 40 + 2 * half;
  const float* wc = Wtl + (size_t)l16 * 60 + 2 * half;
  v8f acc = {};
  for (int k = 0; k < 60; k += 4) {
    int ka = k + 2 * half;                            // even; never straddles 20/40
    const float* bp = (ka < 20) ? (r1 + k) : (ka < 40) ? (r2 + k) : (r3 + k);
    v2f a = *(const v2f*)bp;
    v2f b = *(const v2f*)(wc + k);                    // unguarded, zero-padded cols
    acc = __builtin_amdgcn_wmma_f32_16x16x4_f32(false, a, false, b,
                                                (short)0, acc, false, false);
  }
  if (l16 < CD) {
    int base = ((wave << 4) + (half << 3)) * CD + l16;
    float bb = bl[l16];
    #pragma unroll
    for (int v = 0; v < 8; v++) out[base + v * CD] = acc[v] + bb;
  }
}

// ---------------------------------------------------------------- launcher
extern "C" void kernel_launch(void* const* d_in, const int* in_sizes, int n_in,
                              void* d_out, int out_size, void* d_ws, size_t ws_size,
                              hipStream_t stream) {
  const float* x   = (const float*)d_in[0];
  const int*   ei  = (const int*)d_in[1];
  const float* ew  = (const float*)d_in[2];
  const float* W1  = (const float*)d_in[3];
  const float* b1  = (const float*)d_in[4];
  const float* g1  = (const float*)d_in[5];
  const float* be1 = (const float*)d_in[6];
  const float* W2  = (const float*)d_in[7];
  const float* b2  = (const float*)d_in[8];
  const float* g2  = (const float*)d_in[9];
  const float* be2 = (const float*)d_in[10];
  const float* W3  = (const float*)d_in[11];
  const float* b3  = (const float*)d_in[12];
  const float* Wl  = (const float*)d_in[13];
  const float* bl  = (const float*)d_in[14];
  float* out = (float*)d_out;

  const int F = 128;
  const int N = in_sizes[0] / F;      // 100000 (multiple of 16)
  const int E = in_sizes[1] / 2;      // 3200000
  const int* srcp = ei;
  const int* dstp = ei + E;

  // workspace layout (floats)
  float* ws    = (float*)d_ws;
  float* dinv  = ws;                  // N
  float* h     = dinv + N;            // N*HD
  float* conv  = h    + N * HD;       // N*HD
  float* o1    = conv + N * HD;       // N*HD
  float* o2    = o1   + N * HD;       // N*HD
  float* o3    = o2   + N * HD;       // N*HD
  float* stats = o3   + N * HD;       // 80
  float* Wt1   = stats + 80;          // HDPAD * 128 = 4096
  float* Wt2   = Wt1 + HDPAD * F;     // HDPAD * 20  = 640
  float* Wt3   = Wt2 + HDPAD * HD;    // HDPAD * 20  = 640
  float* Wtl   = Wt3 + HDPAD * HD;    // CDPAD * 60  = 960

  const int nodeBlk = (N + 255) / 256;
  const int elemBlk = (N * HD + 255) / 256;
  const int edgeBlk = (E + 255) / 256;

  // pack all weight matrices once (zero-padded transpose)
  pack_w_kernel<<<(HDPAD * F  + 255) / 256, 256, 0, stream>>>(W1, Wt1, F,  HD, HDPAD);
  pack_w_kernel<<<(HDPAD * HD + 255) / 256, 256, 0, stream>>>(W2, Wt2, HD, HD, HDPAD);
  pack_w_kernel<<<(HDPAD * HD + 255) / 256, 256, 0, stream>>>(W3, Wt3, HD, HD, HDPAD);
  pack_w_kernel<<<(CDPAD * 60 + 255) / 256, 256, 0, stream>>>(Wl, Wtl, 60, CD, CDPAD);

  // degree -> dinv
  fill_zero_kernel<<<nodeBlk, 256, 0, stream>>>(dinv, N);
  deg_kernel<<<edgeBlk, 256, 0, stream>>>(dstp, ew, dinv, E);
  dinv_kernel<<<nodeBlk, 256, 0, stream>>>(dinv, N);

  auto gemm = [&](const float* X, int K, const float* Wt, float* Out) {
    int waves  = (N / 16) * (HDPAD / 16);
    int blocks = (waves * 32 + 255) / 256;
    gemm_wmma_kernel<<<blocks, 256, 0, stream>>>(X, Wt, Out, N, K, HD, HDPAD);
  };

  auto layer = [&](const float* X, int K, const float* Wt, const float* bias,
                   const float* g, const float* be, float* o, bool bn) {
    gemm(X, K, Wt, h);
    fill_zero_kernel<<<elemBlk, 256, 0, stream>>>(conv, N * HD);
    edge_scatter_kernel<<<edgeBlk, 256, 0, stream>>>(srcp, dstp, ew, dinv, h, conv, E);
    if (bn) fill_zero_kernel<<<1, 64, 0, stream>>>(stats, 2 * HD);
    post_kernel<<<nodeBlk, 256, 0, stream>>>(conv, h, dinv, bias, o, stats, N, bn ? 1 : 0);
    if (bn) {
      bn_finalize_kernel<<<1, 32, 0, stream>>>(stats, g, be, 1.0f / (float)N);
      bn_apply_kernel<<<elemBlk, 256, 0, stream>>>(o, stats, N * HD);
    }
  };

  layer(x,  F,  Wt1, b1, g1, be1, o1, true);
  layer(o1, HD, Wt2, b2, g2, be2, o2, true);
  layer(o2, HD, Wt3, b3, nullptr, nullptr, o3, false);

  int fwaves  = N / 16;
  int fblocks = (fwaves * 32 + 255) / 256;
  final_gemm_kernel<<<fblocks, 256, 0, stream>>>(o1, o2, o3, Wtl, bl, out, N);
}